// PastFutureContext_2362232012961
// MI455X (gfx1250) — compile-verified
//
#include <hip/hip_runtime.h>
#include <hip/hip_bf16.h>

typedef __attribute__((ext_vector_type(2))) float v2f;
typedef __attribute__((ext_vector_type(8))) float v8f;
typedef __attribute__((ext_vector_type(4))) unsigned int v4u;
typedef __attribute__((ext_vector_type(8))) int v8i;
typedef __attribute__((ext_vector_type(4))) int v4i;

#define BATCH 8
#define SEQ   1024
#define DD    512      // d == h == 512
#define K3TOT 1536

static __device__ __forceinline__ v8f wmma_f32(v2f a, v2f b, v8f c) {
  // V_WMMA_F32_16X16X4_F32 : D = A(16x4,f32) * B(4x16,f32) + C(16x16,f32)
  return __builtin_amdgcn_wmma_f32_16x16x4_f32(
      /*neg_a=*/false, a, /*neg_b=*/false, b,
      /*c_mod=*/(short)0, c, /*reuse_a=*/false, /*reuse_b=*/false);
}

// ---------------------------------------------------------------------------
// TDM: 2D tile load, global -> LDS, with LDS row padding.
// Tile: tile_m rows x tile_k elems (f32) of a row-major tensor whose row
// stride is row_stride elems. LDS gets pad_amt DWORDs of pad every
// pad_ivl DWORDs (codes per D# group1 encoding; pad row stride).
// ---------------------------------------------------------------------------
static __device__ __forceinline__ void tdm_load_2d(
    unsigned lds_off, const float* gptr,
    unsigned tile_k, unsigned tile_m,
    unsigned tensor_k, unsigned tensor_m,
    unsigned row_stride,
    unsigned pad_en, unsigned pad_ivl_code, unsigned pad_amt_code) {
  unsigned long long ga = (unsigned long long)(uintptr_t)gptr;
  v4u g0;
  g0[0] = 1u;                                            // count=1 (valid user D#)
  g0[1] = lds_off;                                       // lds_addr (bytes)
  g0[2] = (unsigned)(ga & 0xFFFFFFFFu);                  // global_addr[31:0]
  g0[3] = (unsigned)((ga >> 32) & 0x01FFFFFFu)           // global_addr[56:32]
          | (2u << 30);                                  // type = 2 ("image")
  v8i g1;
  g1[0] = (int)((2u << 16)                               // data_size = 4B
                | (pad_en << 20)
                | (pad_ivl_code << 22)
                | (pad_amt_code << 25));
  g1[1] = (int)((tensor_k & 0xFFFFu) << 16);             // tensor_dim0[15:0]
  g1[2] = (int)(((tensor_k >> 16) & 0xFFFFu)             // tensor_dim0[31:16]
                | ((tensor_m & 0xFFFFu) << 16));         // tensor_dim1[15:0]
  g1[3] = (int)(((tensor_m >> 16) & 0xFFFFu)             // tensor_dim1[31:16]
                | ((tile_k & 0xFFFFu) << 16));           // tile_dim0
  g1[4] = (int)(tile_m & 0xFFFFu);                       // tile_dim1 (tile_dim2=0)
  g1[5] = (int)row_stride;                               // tensor_dim0_stride[31:0]
  g1[6] = 0;                                             // stride[47:32], dim1_stride lo
  g1[7] = 0;
  v4i gz = {0, 0, 0, 0};
#if __clang_major__ >= 23
  v8i gz8 = {0, 0, 0, 0, 0, 0, 0, 0};
  __builtin_amdgcn_tensor_load_to_lds(g0, g1, gz, gz, gz8, 0);
#else
  __builtin_amdgcn_tensor_load_to_lds(g0, g1, gz, gz, 0);
#endif
}

// Per-lane async copy of 16B global -> LDS (tracked on ASYNCcnt).
static __device__ __forceinline__ void async_cp16(unsigned lds_off,
                                                  const float* g) {
  asm volatile("global_load_async_to_lds_b128 %0, %1, off"
               :: "v"(lds_off), "v"(g) : "memory");
}
static __device__ __forceinline__ void async_wait0() {
  asm volatile("s_wait_asynccnt 0x0" ::: "memory");
}

// ---------------------------------------------------------------------------
// K0: per-batch prefix scan of the mask -> row scales sm_i/(cnt+eps)
// ---------------------------------------------------------------------------
__global__ void k_scan(const int* __restrict__ sm,
                       float* __restrict__ invp, float* __restrict__ invf) {
  int b = threadIdx.x;
  if (b >= BATCH) return;
  const int* m = sm + b * SEQ;
  int total = 0;
  for (int j = 0; j < SEQ; ++j) total += m[j];
  int cnt = 0;
  for (int i = 0; i < SEQ; ++i) {
    int mi = m[i];
    int cf = total - cnt - mi;
    invp[b * SEQ + i] = mi ? 1.0f / ((float)cnt + 1e-8f) : 0.0f;
    invf[b * SEQ + i] = mi ? 1.0f / ((float)cf  + 1e-8f) : 0.0f;
    cnt += mi;
  }
}

// ---------------------------------------------------------------------------
// K1: C[M=8192][512] = A[M][512] @ W[512][512]^T + bias   (proj GEMM)
// 64x64 tile per workgroup, 4 waves; A tile staged by TDM, W tile by VALU
// transpose-interleave.
// ---------------------------------------------------------------------------
__global__ __launch_bounds__(128) void k_gemm_nt(
    const float* __restrict__ A, const float* __restrict__ W,
    const float* __restrict__ bias, float* __restrict__ C) {
  __shared__ float As[64 * 72];   // [m][k], row stride 72 (64 + 8 pad)
  __shared__ float Bs[64 * 64];   // interleaved: (k>>1)*128 + n*2 + (k&1)
  const int tid  = threadIdx.x;
  const int w    = tid >> 5;
  const int lane = tid & 31;
  const int lm   = lane & 15;
  const int kh   = lane >> 4;
  const int koff = kh * 2;
  const int m0 = blockIdx.x * 64;
  const int n0 = blockIdx.y * 64;
  const unsigned as_base = (unsigned)(uintptr_t)As;

  v8f c[4] = {};
  for (int k0 = 0; k0 < DD; k0 += 64) {
    if (w == 0) {   // one wave drives the TDM; pad 8 DW every 64 DW -> stride 72
      tdm_load_2d(as_base, A + (size_t)m0 * DD + k0,
                  64, 64, DD, 8192, DD, 1u, 5u, 7u);
    }
    for (int idx = tid; idx < 64 * 16; idx += 128) {   // W tile, transpose+interleave
      int n = idx >> 4, c4 = idx & 15;
      const float4 v = *(const float4*)(W + (size_t)(n0 + n) * DD + k0 + c4 * 4);
      int kb = c4 * 4;
      Bs[(kb >> 1) * 128 + n * 2 + 0]       = v.x;
      Bs[(kb >> 1) * 128 + n * 2 + 1]       = v.y;
      Bs[((kb >> 1) + 1) * 128 + n * 2 + 0] = v.z;
      Bs[((kb >> 1) + 1) * 128 + n * 2 + 1] = v.w;
    }
    if (w == 0) __builtin_amdgcn_s_wait_tensorcnt(0);
    __syncthreads();
#pragma unroll
    for (int kk = 0; kk < 64; kk += 4) {
      const int ka = kk + koff;
      v2f a;
      a.x = As[(w * 16 + lm) * 72 + ka];
      a.y = As[(w * 16 + lm) * 72 + ka + 1];
      const int bb = (ka >> 1) * 128;
#pragma unroll
      for (int t = 0; t < 4; ++t) {
        v2f bf;
        bf.x = Bs[bb + (t * 16 + lm) * 2];
        bf.y = Bs[bb + (t * 16 + lm) * 2 + 1];
        c[t] = wmma_f32(a, bf, c[t]);
      }
    }
    __syncthreads();
  }
#pragma unroll
  for (int t = 0; t < 4; ++t) {
    const int ng = n0 + t * 16 + lm;
    const float bv = bias[ng];
#pragma unroll
    for (int r = 0; r < 8; ++r) {
      const int mg = m0 + w * 16 + kh * 8 + r;
      C[(size_t)mg * DD + ng] = c[t][r] + bv;
    }
  }
}

// ---------------------------------------------------------------------------
// K2: fused  S = proj_tile @ feat^T ; mask ; past/fut = S' @ feat
// One workgroup per (batch, 16-row tile). 8 waves. Wave w:
//   phase1: S columns [w*16, w*16+16)     phase2: output cols [w*64, w*64+64)
// ---------------------------------------------------------------------------
__global__ __launch_bounds__(256) void k_ctx(
    const float* __restrict__ proj, const float* __restrict__ feat,
    const int* __restrict__ sm, const float* __restrict__ invp,
    const float* __restrict__ invf,
    float* __restrict__ past, float* __restrict__ fut) {
  __shared__ float Ps[16 * 516];   // proj row-tile [i][k], stride 516
  __shared__ float Fs[16 * 256];   // feat^T chunk: (k>>1)*256 + j*2 + (k&1), k<16, j<128
  __shared__ float Ss[16 * 132];   // masked S [i][j], stride 132

  const int tid  = threadIdx.x;
  const int w    = tid >> 5;
  const int lane = tid & 31;
  const int lm   = lane & 15;
  const int kh   = lane >> 4;
  const int koff = kh * 2;
  const int b  = blockIdx.y;
  const int i0 = blockIdx.x * 16;
  const float* featB = feat + (size_t)b * SEQ * DD;

  // preload proj tile 16x512 via per-lane async copies (ASYNCcnt path)
  for (int idx = tid; idx < 16 * 128; idx += 256) {
    int r = idx >> 7, c4 = idx & 127;
    async_cp16((unsigned)(uintptr_t)&Ps[r * 516 + c4 * 4],
               proj + ((size_t)(b * SEQ) + i0 + r) * DD + c4 * 4);
  }
  async_wait0();
  __syncthreads();

  v8f pc[4] = {};
  v8f fc[4] = {};

  for (int j0 = 0; j0 < SEQ; j0 += 128) {
    // ---- phase 1: S(16 x 128) = proj_tile(16x512) @ feat_jblock^T ----
    v8f s = {};
    for (int k0 = 0; k0 < DD; k0 += 16) {
      __syncthreads();                                // Fs consumers of prev chunk done
      for (int idx = tid; idx < 128 * 4; idx += 256) {
        int j = idx >> 2, c4 = idx & 3;
        const float4 v = *(const float4*)(featB + (size_t)(j0 + j) * DD + k0 + c4 * 4);
        int kb = c4 * 4;
        Fs[(kb >> 1) * 256 + j * 2 + 0]       = v.x;
        Fs[(kb >> 1) * 256 + j * 2 + 1]       = v.y;
        Fs[((kb >> 1) + 1) * 256 + j * 2 + 0] = v.z;
        Fs[((kb >> 1) + 1) * 256 + j * 2 + 1] = v.w;
      }
      __syncthreads();
#pragma unroll
      for (int kk = 0; kk < 16; kk += 4) {
        const int ka = kk + koff;
        v2f a;
        a.x = Ps[lm * 516 + k0 + ka];
        a.y = Ps[lm * 516 + k0 + ka + 1];
        v2f bf;
        const int bb = (ka >> 1) * 256 + (w * 16 + lm) * 2;
        bf.x = Fs[bb];
        bf.y = Fs[bb + 1];
        s = wmma_f32(a, bf, s);
      }
    }
    // ---- write S * sm_j into LDS ----
    {
      const int jg = j0 + w * 16 + lm;
      const float smj = (float)sm[b * SEQ + jg];
      __syncthreads();                                // prev phase2 readers of Ss done
#pragma unroll
      for (int r = 0; r < 8; ++r)
        Ss[(kh * 8 + r) * 132 + w * 16 + lm] = s[r] * smj;
      __syncthreads();
    }
    // ---- phase 2: past/fut accumulators += S'(16x128) @ feat(128x512) ----
    const int ig = i0 + lm;                           // A-matrix row index
#pragma unroll 4
    for (int jj = 0; jj < 128; jj += 4) {
      const int ja = jj + koff;
      const float s0 = Ss[lm * 132 + ja];
      const float s1 = Ss[lm * 132 + ja + 1];
      const int jg0 = j0 + ja, jg1 = jg0 + 1;
      v2f ap, af;
      ap.x = (jg0 < ig) ? s0 : 0.0f;
      ap.y = (jg1 < ig) ? s1 : 0.0f;
      af.x = (jg0 > ig) ? s0 : 0.0f;
      af.y = (jg1 > ig) ? s1 : 0.0f;
#pragma unroll
      for (int t = 0; t < 4; ++t) {
        const int ng = w * 64 + t * 16 + lm;
        v2f bf;
        bf.x = featB[(size_t)jg0 * DD + ng];
        bf.y = featB[(size_t)jg1 * DD + ng];
        pc[t] = wmma_f32(ap, bf, pc[t]);
        fc[t] = wmma_f32(af, bf, fc[t]);
      }
    }
  }
  // ---- epilogue: apply row scales sm_i/(cnt+eps) ----
#pragma unroll
  for (int r = 0; r < 8; ++r) {
    const int ir = i0 + kh * 8 + r;
    const float sp = invp[b * SEQ + ir];
    const float sf = invf[b * SEQ + ir];
#pragma unroll
    for (int t = 0; t < 4; ++t) {
      const int ng = w * 64 + t * 16 + lm;
      const size_t o = ((size_t)(b * SEQ) + ir) * DD + ng;
      past[o] = pc[t][r] * sp;
      fut[o]  = fc[t][r] * sf;
    }
  }
}

// ---------------------------------------------------------------------------
// K3: out = [past | feat | fut] @ W2[512][1536]^T + b2   (K=1536)
// ---------------------------------------------------------------------------
__global__ __launch_bounds__(128) void k_gemm_out(
    const float* __restrict__ past, const float* __restrict__ feat,
    const float* __restrict__ fut, const float* __restrict__ W2,
    const float* __restrict__ b2, float* __restrict__ out) {
  __shared__ float As[64 * 72];
  __shared__ float Bs[64 * 64];
  const int tid  = threadIdx.x;
  const int w    = tid >> 5;
  const int lane = tid & 31;
  const int lm   = lane & 15;
  const int kh   = lane >> 4;
  const int koff = kh * 2;
  const int m0 = blockIdx.x * 64;
  const int n0 = blockIdx.y * 64;
  const unsigned as_base = (unsigned)(uintptr_t)As;

  v8f c[4] = {};
  for (int k0 = 0; k0 < K3TOT; k0 += 64) {
    const float* src = (k0 < DD) ? past : (k0 < 2 * DD) ? feat : fut;
    const int kl = k0 & (DD - 1);
    if (w == 0) {
      tdm_load_2d(as_base, src + (size_t)m0 * DD + kl,
                  64, 64, DD, 8192, DD, 1u, 5u, 7u);
    }
    for (int idx = tid; idx < 64 * 16; idx += 128) {
      int n = idx >> 4, c4 = idx & 15;
      const float4 v = *(const float4*)(W2 + (size_t)(n0 + n) * K3TOT + k0 + c4 * 4);
      int kb = c4 * 4;
      Bs[(kb >> 1) * 128 + n * 2 + 0]       = v.x;
      Bs[(kb >> 1) * 128 + n * 2 + 1]       = v.y;
      Bs[((kb >> 1) + 1) * 128 + n * 2 + 0] = v.z;
      Bs[((kb >> 1) + 1) * 128 + n * 2 + 1] = v.w;
    }
    if (w == 0) __builtin_amdgcn_s_wait_tensorcnt(0);
    __syncthreads();
#pragma unroll
    for (int kk = 0; kk < 64; kk += 4) {
      const int ka = kk + koff;
      v2f a;
      a.x = As[(w * 16 + lm) * 72 + ka];
      a.y = As[(w * 16 + lm) * 72 + ka + 1];
      const int bb = (ka >> 1) * 128;
#pragma unroll
      for (int t = 0; t < 4; ++t) {
        v2f bf;
        bf.x = Bs[bb + (t * 16 + lm) * 2];
        bf.y = Bs[bb + (t * 16 + lm) * 2 + 1];
        c[t] = wmma_f32(a, bf, c[t]);
      }
    }
    __syncthreads();
  }
#pragma unroll
  for (int t = 0; t < 4; ++t) {
    const int ng = n0 + t * 16 + lm;
    const float bv = b2[ng];
#pragma unroll
    for (int r = 0; r < 8; ++r) {
      const int mg = m0 + w * 16 + kh * 8 + r;
      out[(size_t)mg * DD + ng] = c[t][r] + bv;
    }
  }
}

// ---------------------------------------------------------------------------
extern "C" void kernel_launch(void* const* d_in, const int* in_sizes, int n_in,
                              void* d_out, int out_size, void* d_ws, size_t ws_size,
                              hipStream_t stream) {
  const float* features    = (const float*)d_in[0];
  const int*   states_mask = (const int*)d_in[1];
  const float* W1 = (const float*)d_in[2];
  const float* b1 = (const float*)d_in[3];
  const float* W2 = (const float*)d_in[4];
  const float* b2 = (const float*)d_in[5];
  float* out = (float*)d_out;
  float* ws  = (float*)d_ws;

  const size_t FS = (size_t)BATCH * SEQ * DD;   // 4,194,304 floats per tensor
  float* proj = ws;
  float* past = ws + FS;
  float* fut  = ws + 2 * FS;
  float* invp = ws + 3 * FS;
  float* invf = invp + BATCH * SEQ;

  k_scan<<<1, 32, 0, stream>>>(states_mask, invp, invf);
  k_gemm_nt<<<dim3(128, 8), 128, 0, stream>>>(features, W1, b1, proj);
  k_ctx<<<dim3(SEQ / 16, BATCH), 256, 0, stream>>>(proj, features, states_mask,
                                                   invp, invf, past, fut);
  k_gemm_out<<<dim3(128, 8), 128, 0, stream>>>(past, features, fut, W2, b2, out);
}